// GNN_19825569038772
// MI455X (gfx1250) — compile-verified
//
#include <hip/hip_runtime.h>

// ---------------------------------------------------------------------------
// GNN pipeline for MI455X (gfx1250, wave32).
//   GraphNorm -> GraphConv(+ReLU) x3 -> mean-pool -> MLP head -> softmax
// GEMMs use V_WMMA_F32_16X16X4_F32 (native fp32 WMMA).
// Edge scatter + segment stats use global fp32 atomics (L2-resident targets).
// ---------------------------------------------------------------------------

typedef float v2f __attribute__((ext_vector_type(2)));
typedef float v8f __attribute__((ext_vector_type(8)));

#define N_NODES  100000
#define N_EDGES  3200000
#define N_GRAPHS 1024
#define GN_EPS   1e-5f

// ----------------------------- utility kernels -----------------------------

__global__ void k_zero(float* __restrict__ p, int n) {
  int i = blockIdx.x * blockDim.x + threadIdx.x;
  if (i < n) p[i] = 0.0f;
}

__global__ void k_count(const int* __restrict__ batch, float* __restrict__ cnt, int n) {
  int i = blockIdx.x * blockDim.x + threadIdx.x;
  if (i < n) atomicAdd(&cnt[batch[i]], 1.0f);
}

// sum[g*F+f] += x[i*F+f]
template <int F>
__global__ void k_seg_sum(const float* __restrict__ x, const int* __restrict__ batch,
                          float* __restrict__ sum, int n) {
  int idx = blockIdx.x * blockDim.x + threadIdx.x;
  if (idx >= n * F) return;
  int i = idx / F, f = idx % F;
  atomicAdd(&sum[batch[i] * F + f], x[idx]);
}

// p[i] /= max(cnt[i/F], 1)
__global__ void k_div_cnt(float* __restrict__ p, const float* __restrict__ cnt,
                          int gf, int F) {
  int i = blockIdx.x * blockDim.x + threadIdx.x;
  if (i >= gf) return;
  p[i] = p[i] / fmaxf(cnt[i / F], 1.0f);
}

// t = x - mean[g]*ms ; out = t ; var[g] += t*t
template <int F>
__global__ void k_center(const float* __restrict__ x, const int* __restrict__ batch,
                         const float* __restrict__ mean, const float* __restrict__ ms,
                         float* __restrict__ out, float* __restrict__ var, int n) {
  int idx = blockIdx.x * blockDim.x + threadIdx.x;
  if (idx >= n * F) return;
  int i = idx / F, f = idx % F;
  int g = batch[i];
  float t = x[idx] - mean[g * F + f] * ms[f];
  out[idx] = t;
  atomicAdd(&var[g * F + f], t * t);
}

// var -> rsqrt(var/cnt + eps)
__global__ void k_invstd(float* __restrict__ var, const float* __restrict__ cnt,
                         int gf, int F) {
  int i = blockIdx.x * blockDim.x + threadIdx.x;
  if (i >= gf) return;
  float v = var[i] / fmaxf(cnt[i / F], 1.0f);
  var[i] = rsqrtf(v + GN_EPS);
}

// out = w * (out * invstd[g]) + b   (in place on centered values)
template <int F>
__global__ void k_scale(float* __restrict__ out, const int* __restrict__ batch,
                        const float* __restrict__ invstd, const float* __restrict__ w,
                        const float* __restrict__ b, int n) {
  int idx = blockIdx.x * blockDim.x + threadIdx.x;
  if (idx >= n * F) return;
  int i = idx / F, f = idx % F;
  int g = batch[i];
  out[idx] = w[f] * (out[idx] * invstd[g * F + f]) + b[f];
}

// agg[dst[e]*F+f] += h[src[e]*F+f]
template <int F>
__global__ void k_scatter(const float* __restrict__ h, const int* __restrict__ src,
                          const int* __restrict__ dst, float* __restrict__ agg, int e) {
  int idx = blockIdx.x * blockDim.x + threadIdx.x;
  if (idx >= e * F) return;
  int ee = idx / F, f = idx % F;
  int s = src[ee], d = dst[ee];
  atomicAdd(&agg[d * F + f], h[s * F + f]);
}

// ------------------------- WMMA GraphConv GEMM -----------------------------
// out[N,64] = relu( agg[N,KIN] @ w_rel^T + bias + xin[N,KIN] @ w_root^T )
// One wave per 16-node tile; 4 accumulator tiles cover the 64 output cols.
// Fragment layouts per CDNA5 ISA 7.12.2 (f32 16x16x4):
//   A: lane l (r=l&15, hi=l>>4) holds {A[r][k0+2hi], A[r][k0+2hi+1]}
//   B: lane l holds {B[k0+2hi][c], B[k0+2hi+1][c]} with c=l&15
//   C/D: vgpr j -> row j+8*hi, col l&15
template <int KIN>
__global__ void conv_wmma_kernel(const float* __restrict__ agg,
                                 const float* __restrict__ xin,
                                 const float* __restrict__ w_rel,   // [64,KIN]
                                 const float* __restrict__ w_root,  // [64,KIN]
                                 const float* __restrict__ bias,    // [64]
                                 float* __restrict__ out,           // [N,64]
                                 int n_nodes) {
  const int lane = threadIdx.x & 31;
  const int wave = threadIdx.x >> 5;
  const int wavesPerBlock = blockDim.x >> 5;
  const int tile = blockIdx.x * wavesPerBlock + wave;
  const int row0 = tile * 16;
  if (row0 >= n_nodes) return;  // whole-wave uniform exit (EXEC all-1 for WMMA)

  const int r = lane & 15;
  const int hi = lane >> 4;

  v8f acc[4] = {};  // 4 x (16x16) f32 accumulators -> columns 0..63

  const float* arow = agg + (size_t)(row0 + r) * KIN;
  const float* xrow = xin + (size_t)(row0 + r) * KIN;

#pragma unroll
  for (int k0 = 0; k0 < KIN; k0 += 4) {
    const int ka = k0 + 2 * hi;
    v2f a = *(const v2f*)(arow + ka);
#pragma unroll
    for (int ot = 0; ot < 4; ++ot) {
      const int col = ot * 16 + r;
      v2f b = *(const v2f*)(w_rel + (size_t)col * KIN + ka);
      acc[ot] = __builtin_amdgcn_wmma_f32_16x16x4_f32(
          false, a, false, b, (short)0, acc[ot], false, false);
    }
  }
#pragma unroll
  for (int k0 = 0; k0 < KIN; k0 += 4) {
    const int ka = k0 + 2 * hi;
    v2f a = *(const v2f*)(xrow + ka);
#pragma unroll
    for (int ot = 0; ot < 4; ++ot) {
      const int col = ot * 16 + r;
      v2f b = *(const v2f*)(w_root + (size_t)col * KIN + ka);
      acc[ot] = __builtin_amdgcn_wmma_f32_16x16x4_f32(
          false, a, false, b, (short)0, acc[ot], false, false);
    }
  }

#pragma unroll
  for (int ot = 0; ot < 4; ++ot) {
    const int col = ot * 16 + r;
    const float bv = bias[col];
#pragma unroll
    for (int j = 0; j < 8; ++j) {
      int row = row0 + j + 8 * hi;
      float v = acc[ot][j] + bv;
      v = v > 0.0f ? v : 0.0f;
      if (row < n_nodes) out[(size_t)row * 64 + col] = v;
    }
  }
}

// ------------------------------- MLP head ----------------------------------
// per graph: relu(pooled @ dense_w^T + db) @ out_w^T + ob -> softmax(2)
__global__ void k_head(const float* __restrict__ pooled, const float* __restrict__ dw,
                       const float* __restrict__ db, const float* __restrict__ ow,
                       const float* __restrict__ ob, float* __restrict__ out) {
  __shared__ float sp[64];
  __shared__ float sh[64];
  const int g = blockIdx.x;
  const int t = threadIdx.x;  // 64 threads
  sp[t] = pooled[g * 64 + t];
  __syncthreads();
  float s = db[t];
#pragma unroll
  for (int k = 0; k < 64; ++k) s += sp[k] * dw[t * 64 + k];
  sh[t] = fmaxf(s, 0.0f);
  __syncthreads();
  if (t < 2) {
    float l = ob[t];
#pragma unroll
    for (int k = 0; k < 64; ++k) l += sh[k] * ow[t * 64 + k];
    sp[t] = l;
  }
  __syncthreads();
  if (t == 0) {
    float l0 = sp[0], l1 = sp[1];
    float m = fmaxf(l0, l1);
    float e0 = __expf(l0 - m), e1 = __expf(l1 - m);
    float inv = 1.0f / (e0 + e1);
    out[g * 2 + 0] = e0 * inv;
    out[g * 2 + 1] = e1 * inv;
  }
}

// ------------------------------ host driver --------------------------------

static inline int cdiv(long a, int b) { return (int)((a + b - 1) / b); }

template <int F>
static void run_graph_norm(const float* xin, float* xout, const int* batch,
                           const float* w, const float* b, const float* ms,
                           float* ssum, float* svar, const float* cnt,
                           hipStream_t stream) {
  const int gf = N_GRAPHS * F;
  const long nf = (long)N_NODES * F;
  k_zero<<<cdiv(gf, 256), 256, 0, stream>>>(ssum, gf);
  k_zero<<<cdiv(gf, 256), 256, 0, stream>>>(svar, gf);
  k_seg_sum<F><<<cdiv(nf, 256), 256, 0, stream>>>(xin, batch, ssum, N_NODES);
  k_div_cnt<<<cdiv(gf, 256), 256, 0, stream>>>(ssum, cnt, gf, F);  // -> mean
  k_center<F><<<cdiv(nf, 256), 256, 0, stream>>>(xin, batch, ssum, ms, xout, svar, N_NODES);
  k_invstd<<<cdiv(gf, 256), 256, 0, stream>>>(svar, cnt, gf, F);
  k_scale<F><<<cdiv(nf, 256), 256, 0, stream>>>(xout, batch, svar, w, b, N_NODES);
}

extern "C" void kernel_launch(void* const* d_in, const int* in_sizes, int n_in,
                              void* d_out, int out_size, void* d_ws, size_t ws_size,
                              hipStream_t stream) {
  const float* x      = (const float*)d_in[0];
  const int*   ei     = (const int*)d_in[1];
  const int*   src    = ei;
  const int*   dst    = ei + N_EDGES;
  const int*   batch  = (const int*)d_in[2];
  const float* gn0_w  = (const float*)d_in[3];
  const float* gn0_b  = (const float*)d_in[4];
  const float* gn0_ms = (const float*)d_in[5];
  const float* gn1_w  = (const float*)d_in[6];
  const float* gn1_b  = (const float*)d_in[7];
  const float* gn1_ms = (const float*)d_in[8];
  const float* gn2_w  = (const float*)d_in[9];
  const float* gn2_b  = (const float*)d_in[10];
  const float* gn2_ms = (const float*)d_in[11];
  const float* w1_rel = (const float*)d_in[12];
  const float* b1     = (const float*)d_in[13];
  const float* w1_root= (const float*)d_in[14];
  const float* w2_rel = (const float*)d_in[15];
  const float* b2     = (const float*)d_in[16];
  const float* w2_root= (const float*)d_in[17];
  const float* w3_rel = (const float*)d_in[18];
  const float* b3     = (const float*)d_in[19];
  const float* w3_root= (const float*)d_in[20];
  const float* dense_w= (const float*)d_in[21];
  const float* dense_b= (const float*)d_in[22];
  const float* out_w  = (const float*)d_in[23];
  const float* out_b  = (const float*)d_in[24];
  float* outp = (float*)d_out;

  // workspace carve-up (floats)
  float* ws   = (float*)d_ws;
  float* h0   = ws;                       // N*16  (gn0 output)
  float* h1   = h0 + (size_t)N_NODES*16;  // N*64
  float* h2   = h1 + (size_t)N_NODES*64;  // N*64
  float* agg  = h2 + (size_t)N_NODES*64;  // N*64 (layer1 uses stride-16 prefix)
  float* cnt  = agg + (size_t)N_NODES*64; // G
  float* ssum = cnt + N_GRAPHS;           // G*64
  float* svar = ssum + (size_t)N_GRAPHS*64; // G*64

  // per-graph node counts (constant across all stages)
  k_zero<<<cdiv(N_GRAPHS, 256), 256, 0, stream>>>(cnt, N_GRAPHS);
  k_count<<<cdiv(N_NODES, 256), 256, 0, stream>>>(batch, cnt, N_NODES);

  const int convBlocks = cdiv((N_NODES + 15) / 16, 8);  // 8 waves / 256-thread block

  // ---- layer 1: gn0 (F=16) -> scatter(16) -> conv 16->64 ----
  run_graph_norm<16>(x, h0, batch, gn0_w, gn0_b, gn0_ms, ssum, svar, cnt, stream);
  k_zero<<<cdiv((long)N_NODES*16, 256), 256, 0, stream>>>(agg, N_NODES*16);
  k_scatter<16><<<cdiv((long)N_EDGES*16, 256), 256, 0, stream>>>(h0, src, dst, agg, N_EDGES);
  conv_wmma_kernel<16><<<convBlocks, 256, 0, stream>>>(agg, h0, w1_rel, w1_root, b1, h1, N_NODES);

  // ---- layer 2: gn1 -> scatter(64) -> conv 64->64 ----
  run_graph_norm<64>(h1, h2, batch, gn1_w, gn1_b, gn1_ms, ssum, svar, cnt, stream);
  k_zero<<<cdiv((long)N_NODES*64, 256), 256, 0, stream>>>(agg, N_NODES*64);
  k_scatter<64><<<cdiv((long)N_EDGES*64, 256), 256, 0, stream>>>(h2, src, dst, agg, N_EDGES);
  conv_wmma_kernel<64><<<convBlocks, 256, 0, stream>>>(agg, h2, w2_rel, w2_root, b2, h1, N_NODES);

  // ---- layer 3: gn2 -> scatter(64) -> conv 64->64 ----
  run_graph_norm<64>(h1, h2, batch, gn2_w, gn2_b, gn2_ms, ssum, svar, cnt, stream);
  k_zero<<<cdiv((long)N_NODES*64, 256), 256, 0, stream>>>(agg, N_NODES*64);
  k_scatter<64><<<cdiv((long)N_EDGES*64, 256), 256, 0, stream>>>(h2, src, dst, agg, N_EDGES);
  conv_wmma_kernel<64><<<convBlocks, 256, 0, stream>>>(agg, h2, w3_rel, w3_root, b3, h1, N_NODES);

  // ---- mean pool -> head ----
  k_zero<<<cdiv(N_GRAPHS*64, 256), 256, 0, stream>>>(ssum, N_GRAPHS*64);
  k_seg_sum<64><<<cdiv((long)N_NODES*64, 256), 256, 0, stream>>>(h1, batch, ssum, N_NODES);
  k_div_cnt<<<cdiv(N_GRAPHS*64, 256), 256, 0, stream>>>(ssum, cnt, N_GRAPHS*64, 64);
  k_head<<<N_GRAPHS, 64, 0, stream>>>(ssum, dense_w, dense_b, out_w, out_b, outp);
}